// APPNP_30897994727892
// MI455X (gfx1250) — compile-verified
//
#include <hip/hip_runtime.h>
#include <math.h>

typedef float v2f __attribute__((ext_vector_type(2)));
typedef float v8f __attribute__((ext_vector_type(8)));

#define ALPHA 0.9f

__device__ __forceinline__ v8f wmma_f32_k4(v2f a, v2f b, v8f c) {
  // V_WMMA_F32_16X16X4_F32: 8-arg form (neg_a, A, neg_b, B, c_mod, C, reuse_a, reuse_b)
  return __builtin_amdgcn_wmma_f32_16x16x4_f32(false, a, false, b, (short)0, c, false, false);
}

// ---------------------------------------------------------------- utilities

__global__ void zero_i32_kernel(int* p, int n) {
  int i = blockIdx.x * blockDim.x + threadIdx.x;
  if (i < n) p[i] = 0;
}

__global__ void count_edges_kernel(const int* __restrict__ dst, int* __restrict__ counts, int E) {
  int i = blockIdx.x * blockDim.x + threadIdx.x;
  if (i < E) atomicAdd(&counts[dst[i]], 1);
}

__global__ void norm_kernel(const int* __restrict__ counts, float* __restrict__ normv, int N) {
  int i = blockIdx.x * blockDim.x + threadIdx.x;
  if (i < N) normv[i] = rsqrtf((float)counts[i] + 1.0f);
}

// Each block sums a 1024-element chunk of counts.
__global__ __launch_bounds__(256) void block_sum_kernel(const int* __restrict__ counts,
                                                        int* __restrict__ blockSums, int N) {
  __shared__ int sm[256];
  int base = blockIdx.x * 1024 + threadIdx.x * 4;
  int s = 0;
#pragma unroll
  for (int j = 0; j < 4; ++j) {
    int idx = base + j;
    if (idx < N) s += counts[idx];
  }
  sm[threadIdx.x] = s;
  __syncthreads();
  for (int off = 128; off > 0; off >>= 1) {
    if (threadIdx.x < off) sm[threadIdx.x] += sm[threadIdx.x + off];
    __syncthreads();
  }
  if (threadIdx.x == 0) blockSums[blockIdx.x] = sm[0];
}

// Tiny sequential exclusive scan over block sums (nb ~ 98).
__global__ void scan_sums_kernel(int* blockSums, int nb) {
  if (blockIdx.x == 0 && threadIdx.x == 0) {
    int run = 0;
    for (int i = 0; i < nb; ++i) {
      int t = blockSums[i];
      blockSums[i] = run;
      run += t;
    }
  }
}

// Exclusive scan of counts -> rowStart (and cursor copy).
__global__ __launch_bounds__(256) void scan_write_kernel(const int* __restrict__ counts,
                                                         const int* __restrict__ blockSums,
                                                         int* __restrict__ rowStart,
                                                         int* __restrict__ cursor, int N) {
  __shared__ int sm[256];
  int t = threadIdx.x;
  int base = blockIdx.x * 1024 + t * 4;
  int c0 = 0, c1 = 0, c2 = 0, c3 = 0;
  if (base + 0 < N) c0 = counts[base + 0];
  if (base + 1 < N) c1 = counts[base + 1];
  if (base + 2 < N) c2 = counts[base + 2];
  if (base + 3 < N) c3 = counts[base + 3];
  int tot = c0 + c1 + c2 + c3;
  sm[t] = tot;
  __syncthreads();
  for (int off = 1; off < 256; off <<= 1) {
    int v = (t >= off) ? sm[t - off] : 0;
    __syncthreads();
    sm[t] += v;
    __syncthreads();
  }
  int excl = sm[t] - tot + blockSums[blockIdx.x];
  int p0 = excl, p1 = p0 + c0, p2 = p1 + c1, p3 = p2 + c2;
  if (base + 0 < N) { rowStart[base + 0] = p0; cursor[base + 0] = p0; }
  if (base + 1 < N) { rowStart[base + 1] = p1; cursor[base + 1] = p1; }
  if (base + 2 < N) { rowStart[base + 2] = p2; cursor[base + 2] = p2; }
  if (base + 3 < N) { rowStart[base + 3] = p3; cursor[base + 3] = p3; }
}

__global__ void scatter_edges_kernel(const int* __restrict__ src, const int* __restrict__ dst,
                                     const float* __restrict__ normv, int* __restrict__ cursor,
                                     int* __restrict__ srcSorted, float* __restrict__ wSorted,
                                     int E) {
  int i = blockIdx.x * blockDim.x + threadIdx.x;
  if (i < E) {
    int s = src[i], d = dst[i];
    int pos = atomicAdd(&cursor[d], 1);
    srcSorted[pos] = s;
    wSorted[pos] = normv[s] * normv[d];
  }
}

// ---------------------------------------------------------------- GEMM 1: [N,F]@[F,64] + bias, relu
// 256 threads = 8 waves; wave w -> rows [blockRow + 16w, +16), all 64 cols (4 WMMA tiles).
__global__ __launch_bounds__(256) void gemm1_wmma_kernel(const float* __restrict__ X,
                                                         const float* __restrict__ W,
                                                         const float* __restrict__ bias,
                                                         float* __restrict__ Y, int N, int F) {
  __shared__ float sA[8 * 2112];  // 8 waves * 16 rows * (128+4) padded cols
  __shared__ float sB[128 * 68];  // 128 K-rows * (64+4) padded cols
  const int w = threadIdx.x >> 5;
  const int lane = threadIdx.x & 31;
  const int m = lane & 15;
  const int hi = lane >> 4;
  const int rowBase = blockIdx.x * 128 + w * 16;
  v8f acc[4] = {};

  for (int k0 = 0; k0 < F; k0 += 128) {
    // stage A chunk: 16 rows x 128 cols per wave, coalesced float4
#pragma unroll
    for (int r = 0; r < 16; ++r) {
      int gr = rowBase + r;
      if (gr >= N) gr = N - 1;
      float4 v = *(const float4*)(X + (size_t)gr * F + k0 + lane * 4);
      *(float4*)(sA + w * 2112 + r * 132 + lane * 4) = v;
    }
    // stage B chunk: 128 x 64 cooperative
#pragma unroll
    for (int j = 0; j < 8; ++j) {
      int idx4 = j * 256 + threadIdx.x;
      int row = idx4 >> 4, c4 = (idx4 & 15) * 4;
      float4 v = *(const float4*)(W + (size_t)(k0 + row) * 64 + c4);
      *(float4*)(sB + row * 68 + c4) = v;
    }
    __syncthreads();

    const float* aPtr = sA + w * 2112 + m * 132 + 2 * hi;
#pragma unroll 8
    for (int ks = 0; ks < 32; ++ks) {
      int kk = ks * 4;
      float2 av = *(const float2*)(aPtr + kk);
      v2f a;
      a[0] = av.x;
      a[1] = av.y;
      const float* b0 = sB + (kk + 2 * hi) * 68 + m;
#pragma unroll
      for (int t = 0; t < 4; ++t) {
        v2f b;
        b[0] = b0[t * 16];
        b[1] = b0[68 + t * 16];
        acc[t] = wmma_f32_k4(a, b, acc[t]);
      }
    }
    __syncthreads();
  }

#pragma unroll
  for (int t = 0; t < 4; ++t) {
    int col = t * 16 + m;
    float bv = bias[col];
#pragma unroll
    for (int i = 0; i < 8; ++i) {
      int gr = rowBase + i + 8 * hi;
      if (gr < N) {
        float v = acc[t][i] + bv;
        Y[(size_t)gr * 64 + col] = v > 0.0f ? v : 0.0f;
      }
    }
  }
}

// ---------------------------------------------------------------- GEMM 2: [N,64]@[64,64] + bias, relu
// Writes both h (ws) and initial r (d_out).
__global__ __launch_bounds__(256) void gemm2_wmma_kernel(const float* __restrict__ A,
                                                         const float* __restrict__ W,
                                                         const float* __restrict__ bias,
                                                         float* __restrict__ Hout,
                                                         float* __restrict__ Rout, int N) {
  __shared__ float sA[8 * 1088];  // 8 waves * 16 rows * 68
  __shared__ float sB[64 * 68];
  const int w = threadIdx.x >> 5;
  const int lane = threadIdx.x & 31;
  const int m = lane & 15;
  const int hi = lane >> 4;
  const int rowBase = blockIdx.x * 128 + w * 16;

#pragma unroll
  for (int j = 0; j < 8; ++j) {
    int idx4 = j * 32 + lane;
    int r = idx4 >> 4, c4 = (idx4 & 15) * 4;
    int gr = rowBase + r;
    if (gr >= N) gr = N - 1;
    float4 v = *(const float4*)(A + (size_t)gr * 64 + c4);
    *(float4*)(sA + w * 1088 + r * 68 + c4) = v;
  }
#pragma unroll
  for (int j = 0; j < 4; ++j) {
    int idx4 = j * 256 + threadIdx.x;
    int row = idx4 >> 4, c4 = (idx4 & 15) * 4;
    float4 v = *(const float4*)(W + (size_t)row * 64 + c4);
    *(float4*)(sB + row * 68 + c4) = v;
  }
  __syncthreads();

  v8f acc[4] = {};
  const float* aPtr = sA + w * 1088 + m * 68 + 2 * hi;
#pragma unroll
  for (int ks = 0; ks < 16; ++ks) {
    int kk = ks * 4;
    float2 av = *(const float2*)(aPtr + kk);
    v2f a;
    a[0] = av.x;
    a[1] = av.y;
    const float* b0 = sB + (kk + 2 * hi) * 68 + m;
#pragma unroll
    for (int t = 0; t < 4; ++t) {
      v2f b;
      b[0] = b0[t * 16];
      b[1] = b0[68 + t * 16];
      acc[t] = wmma_f32_k4(a, b, acc[t]);
    }
  }

#pragma unroll
  for (int t = 0; t < 4; ++t) {
    int col = t * 16 + m;
    float bv = bias[col];
#pragma unroll
    for (int i = 0; i < 8; ++i) {
      int gr = rowBase + i + 8 * hi;
      if (gr < N) {
        float v = acc[t][i] + bv;
        v = v > 0.0f ? v : 0.0f;
        Hout[(size_t)gr * 64 + col] = v;
        Rout[(size_t)gr * 64 + col] = v;
      }
    }
  }
}

// ---------------------------------------------------------------- propagation (CSR gather, no atomics)
// One wave per destination row; lane owns channels (2*lane, 2*lane+1).
__global__ __launch_bounds__(256) void prop_kernel(const float* __restrict__ rIn,
                                                   float* __restrict__ rOut,
                                                   const float* __restrict__ h,
                                                   const float* __restrict__ normv,
                                                   const int* __restrict__ rowStart,
                                                   const int* __restrict__ counts,
                                                   const int* __restrict__ srcS,
                                                   const float* __restrict__ wS, int N) {
  int gid = blockIdx.x * blockDim.x + threadIdx.x;
  int row = gid >> 5;
  int lane = gid & 31;
  if (row >= N) return;
  int start = rowStart[row];
  int cnt = counts[row];
  int c = lane * 2;
  float ax = 0.0f, ay = 0.0f;
  int i = 0;
  for (; i + 1 < cnt; i += 2) {
    int s0 = srcS[start + i];
    int s1 = srcS[start + i + 1];
    float w0 = wS[start + i];
    float w1 = wS[start + i + 1];
    float2 v0 = *(const float2*)(rIn + (size_t)s0 * 64 + c);
    float2 v1 = *(const float2*)(rIn + (size_t)s1 * 64 + c);
    ax += w0 * v0.x + w1 * v1.x;
    ay += w0 * v0.y + w1 * v1.y;
  }
  if (i < cnt) {
    int s0 = srcS[start + i];
    float w0 = wS[start + i];
    float2 v0 = *(const float2*)(rIn + (size_t)s0 * 64 + c);
    ax += w0 * v0.x;
    ay += w0 * v0.y;
  }
  float nv = normv[row];
  float sw = nv * nv;
  float2 rv = *(const float2*)(rIn + (size_t)row * 64 + c);
  float2 hv = *(const float2*)(h + (size_t)row * 64 + c);
  float2 o;
  o.x = ALPHA * (ax + sw * rv.x) + (1.0f - ALPHA) * hv.x;
  o.y = ALPHA * (ay + sw * rv.y) + (1.0f - ALPHA) * hv.y;
  *(float2*)(rOut + (size_t)row * 64 + c) = o;
}

// ---------------------------------------------------------------- softmax over 64 channels, wave per row
__global__ __launch_bounds__(256) void softmax_kernel(float* __restrict__ r, int N) {
  int gid = blockIdx.x * blockDim.x + threadIdx.x;
  int row = gid >> 5;
  int lane = gid & 31;
  if (row >= N) return;
  float2 v = *(const float2*)(r + (size_t)row * 64 + lane * 2);
  float m = fmaxf(v.x, v.y);
  for (int off = 16; off > 0; off >>= 1) m = fmaxf(m, __shfl_xor(m, off, 32));
  float ex = expf(v.x - m);
  float ey = expf(v.y - m);
  float s = ex + ey;
  for (int off = 16; off > 0; off >>= 1) s += __shfl_xor(s, off, 32);
  float inv = 1.0f / s;
  float2 o;
  o.x = ex * inv;
  o.y = ey * inv;
  *(float2*)(r + (size_t)row * 64 + lane * 2) = o;
}

// ---------------------------------------------------------------- launch

extern "C" void kernel_launch(void* const* d_in, const int* in_sizes, int n_in,
                              void* d_out, int out_size, void* d_ws, size_t ws_size,
                              hipStream_t stream) {
  const float* X = (const float*)d_in[0];
  const int* ei = (const int*)d_in[1];
  const float* W1 = (const float*)d_in[2];
  const float* b1 = (const float*)d_in[3];
  const float* W2 = (const float*)d_in[4];
  const float* b2 = (const float*)d_in[5];

  const int H = in_sizes[3];          // 64
  const int F = in_sizes[2] / H;      // 512
  const int N = in_sizes[0] / F;      // 100000
  const int E = in_sizes[1] / 2;      // 3200000
  const int* srcIdx = ei;             // edge_index[0]
  const int* dstIdx = ei + E;         // edge_index[1]

  // workspace carve-out (512B aligned slices)
  char* p = (char*)d_ws;
  auto alloc = [&](size_t bytes) {
    char* r = p;
    p += (bytes + 511) & ~(size_t)511;
    return r;
  };
  int nb = (N + 1023) / 1024;
  int* counts = (int*)alloc((size_t)N * 4);
  int* rowStart = (int*)alloc((size_t)N * 4);
  int* cursor = (int*)alloc((size_t)N * 4);
  float* normv = (float*)alloc((size_t)N * 4);
  int* blockSums = (int*)alloc((size_t)nb * 4);
  int* srcSorted = (int*)alloc((size_t)E * 4);
  float* wSorted = (float*)alloc((size_t)E * 4);
  float* hbuf = (float*)alloc((size_t)N * 64 * 4);
  float* rB = (float*)alloc((size_t)N * 64 * 4);
  float* rA = (float*)d_out;

  // degrees -> norm -> CSR by destination
  zero_i32_kernel<<<(N + 255) / 256, 256, 0, stream>>>(counts, N);
  count_edges_kernel<<<(E + 255) / 256, 256, 0, stream>>>(dstIdx, counts, E);
  norm_kernel<<<(N + 255) / 256, 256, 0, stream>>>(counts, normv, N);
  block_sum_kernel<<<nb, 256, 0, stream>>>(counts, blockSums, N);
  scan_sums_kernel<<<1, 32, 0, stream>>>(blockSums, nb);
  scan_write_kernel<<<nb, 256, 0, stream>>>(counts, blockSums, rowStart, cursor, N);
  scatter_edges_kernel<<<(E + 255) / 256, 256, 0, stream>>>(srcIdx, dstIdx, normv, cursor,
                                                            srcSorted, wSorted, E);

  // MLP: h1 staged in rB, then h -> hbuf and r0 -> d_out
  int gblocks = (N + 127) / 128;
  gemm1_wmma_kernel<<<gblocks, 256, 0, stream>>>(X, W1, b1, rB, N, F);
  gemm2_wmma_kernel<<<gblocks, 256, 0, stream>>>(rB, W2, b2, hbuf, rA, N);

  // APPNP iterations, ping-pong between d_out and rB; ends in d_out after 10 iters
  int propBlocks = (N * 32 + 255) / 256;
  for (int it = 0; it < 10; ++it) {
    const float* rin = (it & 1) ? rB : rA;
    float* rout = (it & 1) ? rA : rB;
    prop_kernel<<<propBlocks, 256, 0, stream>>>(rin, rout, hbuf, normv, rowStart, counts,
                                                srcSorted, wSorted, N);
  }

  softmax_kernel<<<propBlocks, 256, 0, stream>>>(rA, N);
}